// LSTM_84344567759011
// MI455X (gfx1250) — compile-verified
//
#include <hip/hip_runtime.h>
#include <math.h>

// ---------------------------------------------------------------------------
// 3-layer LSTM (B=16384, T=200, F=10, H=30) + FC(30->1), fused over time.
// Per-wave: 16 batch rows for the whole sequence. Gate matmuls use fp32 WMMA
// V_WMMA_F32_16X16X4_F32 (M=16 batch, N=128 padded gates, K=4 slices).
// All weights (incl. folded bias row) live in VGPRs (~688 regs, CDNA5 allows
// 1024/wave); biases are folded into a K-slot whose A value is 1.0.
// ---------------------------------------------------------------------------

typedef __attribute__((ext_vector_type(2))) float v2f;
typedef __attribute__((ext_vector_type(4))) float v4f;
typedef __attribute__((ext_vector_type(8))) float v8f;

#define B_TOT 16384
#define T_LEN 200
#define F_IN  10
#define H_DIM 30
#define G4    120          // 4*H gate columns
#define NT    8            // N tiles of 16 -> 128 padded columns
#define KS1   11           // layer1 k-steps: 3 (x 10 + bias + pad) + 8 (h)
#define KS23  16           // layers 2/3   : 8 (h below + bias slot) + 8 (h self)
#define WAVES 4            // waves per block (128 threads)

#define LOG2E 1.44269504088896340736f

__device__ __forceinline__ float fsigmoid(float x) {
    // 1 / (1 + 2^(-x*log2e))
    return __builtin_amdgcn_rcpf(1.0f + __builtin_amdgcn_exp2f(-x * LOG2E));
}
__device__ __forceinline__ float ftanh(float x) {
    // 1 - 2 / (2^(2x*log2e) + 1)
    return 1.0f - 2.0f * __builtin_amdgcn_rcpf(
                     __builtin_amdgcn_exp2f(x * (2.0f * LOG2E)) + 1.0f);
}

// Stage weights into LDS in WMMA B-operand layout (coalesced one-time pass).
// Slot (nt, ks, lane) = float2 { W(k0, n), W(k0+1, n) },
//   n = nt*16 + (lane&15),  k0 = ks*4 + 2*(lane>>4)
// W(k,n): k<Kin -> Wih[n][k];  k==Kin -> bih[n]+bhh[n] (bias row);
//         koff<=k<koff+30 -> Whh[n][k-koff];  else 0.
__device__ __forceinline__ void fill_wbuf(float* wb,
                                          const float* __restrict__ Wih,
                                          const float* __restrict__ Whh,
                                          const float* __restrict__ bih,
                                          const float* __restrict__ bhh,
                                          int Kin, int koff, int KS,
                                          int tid, int nthreads) {
    const int total = NT * KS * 32;
    for (int idx = tid; idx < total; idx += nthreads) {
        const int lane = idx & 31;
        const int ksnt = idx >> 5;
        const int ks = ksnt % KS;
        const int nt = ksnt / KS;
        const int n  = nt * 16 + (lane & 15);
        const int k0 = ks * 4 + 2 * (lane >> 4);
        float v0 = 0.0f, v1 = 0.0f;
        if (n < G4) {
            int k = k0;
            if (k < Kin)                            v0 = Wih[n * Kin + k];
            else if (k == Kin)                      v0 = bih[n] + bhh[n];
            else if (k >= koff && k < koff + H_DIM) v0 = Whh[n * H_DIM + (k - koff)];
            k = k0 + 1;
            if (k < Kin)                            v1 = Wih[n * Kin + k];
            else if (k == Kin)                      v1 = bih[n] + bhh[n];
            else if (k >= koff && k < koff + H_DIM) v1 = Whh[n * H_DIM + (k - koff)];
        }
        wb[idx * 2 + 0] = v0;
        wb[idx * 2 + 1] = v1;
    }
}

// One LSTM layer step for this wave's 16-row batch tile.
//   KS : total k-steps.  NRA : k-steps from registers (layer-1 x fragment).
//   A order: [registers (NRA)] [ain LDS buf (KS-NRA-8)] [hself LDS buf (8)]
// z (with bias already accumulated) goes to zb column-major (z[n][m]),
// gates applied, new h scattered into hself (A-operand layout).
template <int KS, int NRA>
__device__ __forceinline__ void layer_step(const v2f (&w)[KS * NT],
                                           const v2f* areg, const float* ain,
                                           float* hself, float* zb,
                                           float* c, const int* hoff,
                                           const float* bzp,
                                           int lane, float* zst) {
    constexpr int NIN = KS - NRA - 8;
    v2f a[KS];
#pragma unroll
    for (int ks = 0; ks < NRA; ++ks) a[ks] = areg[ks];
#pragma unroll
    for (int ks = 0; ks < NIN; ++ks)
        a[NRA + ks] = *(const v2f*)(ain + (ks * 32 + lane) * 2);
#pragma unroll
    for (int ks = 0; ks < 8; ++ks)
        a[NRA + NIN + ks] = *(const v2f*)(hself + (ks * 32 + lane) * 2);

#pragma unroll
    for (int nt = 0; nt < NT; ++nt) {
        v8f acc = {0.f, 0.f, 0.f, 0.f, 0.f, 0.f, 0.f, 0.f};
#pragma unroll
        for (int ks = 0; ks < KS; ++ks) {
            acc = __builtin_amdgcn_wmma_f32_16x16x4_f32(
                false, a[ks], false, w[nt * KS + ks], (short)0, acc,
                false, false);
        }
        // column-major store: z[n = nt*16+m16][m = r + 8*hi]; per-lane base
        // zst = zb + m16*16 + 8*hi, so this is base + nt*256 + {0,4} floats.
        v4f lo = __builtin_shufflevector(acc, acc, 0, 1, 2, 3);
        v4f hi4 = __builtin_shufflevector(acc, acc, 4, 5, 6, 7);
        *(v4f*)(zst + nt * 256)     = lo;
        *(v4f*)(zst + nt * 256 + 4) = hi4;
    }

    // Gates: lane handles row m = lane>>1, units j = 15*(lane&1) + r.
    // bzp = zb + j0*16 + mrow ; reads at immediate offsets r*16 + 480*gate.
#pragma unroll
    for (int r = 0; r < 15; ++r) {
        float vi = bzp[r * 16];
        float vf = bzp[r * 16 + 480];
        float vg = bzp[r * 16 + 960];
        float vo = bzp[r * 16 + 1440];
        vi = fsigmoid(vi);
        vf = fsigmoid(vf);
        vg = ftanh(vg);
        vo = fsigmoid(vo);
        const float cc = fmaf(vf, c[r], vi * vg);
        c[r] = cc;
        hself[hoff[r]] = vo * ftanh(cc);
    }
}

__global__ void __launch_bounds__(32 * WAVES, 1)
lstm3_wmma_kernel(const float* __restrict__ x,
                  const float* __restrict__ Wih1, const float* __restrict__ Whh1,
                  const float* __restrict__ bih1, const float* __restrict__ bhh1,
                  const float* __restrict__ Wih2, const float* __restrict__ Whh2,
                  const float* __restrict__ bih2, const float* __restrict__ bhh2,
                  const float* __restrict__ Wih3, const float* __restrict__ Whh3,
                  const float* __restrict__ bih3, const float* __restrict__ bhh3,
                  const float* __restrict__ Wfc,  const float* __restrict__ bfc,
                  float* __restrict__ out) {
    // CDNA5: up to 320 KB LDS per workgroup; this uses ~145 KB.
    __shared__ __align__(64) float smem[NT * KS1 * 64 + 2 * NT * KS23 * 64 +
                                        WAVES * (3 * 512 + 2048)];
    float* wbuf1 = smem;                       // 5632 f (staging only)
    float* wbuf2 = wbuf1 + NT * KS1 * 64;      // 8192 f
    float* wbuf3 = wbuf2 + NT * KS23 * 64;     // 8192 f
    float* wmem  = wbuf3 + NT * KS23 * 64;

    const int tid  = threadIdx.x;
    const int lane = tid & 31;
    const int wave = tid >> 5;

    float* mybase = wmem + wave * (3 * 512 + 2048);
    float* hb1 = mybase;            // 512 f, A-operand layout (8 k-steps)
    float* hb2 = hb1 + 512;
    float* hb3 = hb2 + 512;
    float* zb  = hb3 + 512;         // 128 x 16 f, column-major z[n][m]

    fill_wbuf(wbuf1, Wih1, Whh1, bih1, bhh1, F_IN, 12, KS1, tid, blockDim.x);
    fill_wbuf(wbuf2, Wih2, Whh2, bih2, bhh2, H_DIM, 32, KS23, tid, blockDim.x);
    fill_wbuf(wbuf3, Wih3, Whh3, bih3, bhh3, H_DIM, 32, KS23, tid, blockDim.x);

    // h0 = 0; A-layout slot for k==30 (the layers-2/3 bias row) = 1.0
    for (int i = lane; i < 3 * 512; i += 32) mybase[i] = 0.0f;
    if (lane < 16) {
        hb1[480 + 2 * lane] = 1.0f;   // (ks=7, kr=2, m=lane) -> k==30 slot
        hb2[480 + 2 * lane] = 1.0f;
    }
    __syncthreads();

    // Pull the whole weight set into VGPRs (~688 regs; CDNA5 allows 1024/wave).
    v2f w1[NT * KS1], w2[NT * KS23], w3[NT * KS23];
#pragma unroll
    for (int s = 0; s < NT * KS1; ++s)
        w1[s] = *(const v2f*)(wbuf1 + (s * 32 + lane) * 2);
#pragma unroll
    for (int s = 0; s < NT * KS23; ++s)
        w2[s] = *(const v2f*)(wbuf2 + (s * 32 + lane) * 2);
#pragma unroll
    for (int s = 0; s < NT * KS23; ++s)
        w3[s] = *(const v2f*)(wbuf3 + (s * 32 + lane) * 2);

    float c1[15], c2[15], c3[15];
#pragma unroll
    for (int r = 0; r < 15; ++r) { c1[r] = 0.f; c2[r] = 0.f; c3[r] = 0.f; }

    const int b0   = (blockIdx.x * WAVES + wave) * 16;
    const int m16  = lane & 15;
    const int hi   = lane >> 4;
    const int mrow = lane >> 1;         // gate row handled by this lane
    const int j0   = 15 * (lane & 1);   // first gate unit handled by this lane

    float* zst = zb + m16 * 16 + 8 * hi;          // z-store base (b128 x2 / tile)
    const float* bzp = zb + j0 * 16 + mrow;       // gate-read base (imm offsets)

    // h-scatter offsets (A-operand layout), identical for all three layers.
    int hoff[15];
#pragma unroll
    for (int r = 0; r < 15; ++r) {
        const int j = j0 + r;  // never wraps: j0+14 <= 29
        hoff[r] = ((j >> 2) * 32 + ((j >> 1) & 1) * 16 + mrow) * 2 + (j & 1);
    }

    const float* xrow = x + (size_t)(b0 + m16) * T_LEN * F_IN;

    for (int t = 0; t < T_LEN; ++t) {
        // Layer-1 x fragment (A layout): features 0..9, then bias-1.0, pad 0.
        const float* xr = xrow + (size_t)t * F_IN;
        v2f ax[3];
        ax[0] = *(const v2f*)(xr + 2 * hi);        // k {0,1} | {2,3}
        ax[1] = *(const v2f*)(xr + 4 + 2 * hi);    // k {4,5} | {6,7}
        {
            v2f tld = *(const v2f*)(xr + (hi ? 0 : 8));   // safe addr both ways
            v2f one = {1.0f, 0.0f};                       // k==10 bias slot
            ax[2] = hi ? one : tld;                       // k {8,9} | {1.0, 0}
        }
        layer_step<KS1, 3>(w1, ax, nullptr, hb1, zb, c1, hoff, bzp, lane, zst);
        layer_step<KS23, 0>(w2, nullptr, hb1, hb2, zb, c2, hoff, bzp, lane, zst);
        layer_step<KS23, 0>(w3, nullptr, hb2, hb3, zb, c3, hoff, bzp, lane, zst);
    }

    // FC: out[b] = h3[b] . Wfc + bfc  (lanes 0..15, one batch row each)
    if (lane < 16) {
        float acc = bfc[0];
#pragma unroll
        for (int j = 0; j < H_DIM; ++j) {
            const int ksj = j >> 2, kr = j & 3;
            const float hv = hb3[(ksj * 32 + (kr >> 1) * 16 + lane) * 2 + (kr & 1)];
            acc = fmaf(hv, Wfc[j], acc);
        }
        out[b0 + lane] = acc;
    }
}

extern "C" void kernel_launch(void* const* d_in, const int* in_sizes, int n_in,
                              void* d_out, int out_size, void* d_ws, size_t ws_size,
                              hipStream_t stream) {
    (void)in_sizes; (void)n_in; (void)d_ws; (void)ws_size; (void)out_size;
    const float* x    = (const float*)d_in[0];
    const float* Wih1 = (const float*)d_in[1];
    const float* Whh1 = (const float*)d_in[2];
    const float* bih1 = (const float*)d_in[3];
    const float* bhh1 = (const float*)d_in[4];
    const float* Wih2 = (const float*)d_in[5];
    const float* Whh2 = (const float*)d_in[6];
    const float* bih2 = (const float*)d_in[7];
    const float* bhh2 = (const float*)d_in[8];
    const float* Wih3 = (const float*)d_in[9];
    const float* Whh3 = (const float*)d_in[10];
    const float* bih3 = (const float*)d_in[11];
    const float* bhh3 = (const float*)d_in[12];
    const float* Wfc  = (const float*)d_in[13];
    const float* bfc  = (const float*)d_in[14];
    float* out = (float*)d_out;

    dim3 grid(B_TOT / (16 * WAVES));   // 256 blocks
    dim3 block(32 * WAVES);            // 128 threads = 4 waves
    lstm3_wmma_kernel<<<grid, block, 0, stream>>>(
        x, Wih1, Whh1, bih1, bhh1, Wih2, Whh2, bih2, bhh2,
        Wih3, Whh3, bih3, bhh3, Wfc, bfc, out);
}